// GCN_12781822673280
// MI455X (gfx1250) — compile-verified
//
#include <hip/hip_runtime.h>
#include <hip/hip_bf16.h>

typedef __attribute__((ext_vector_type(2))) float v2f;
typedef __attribute__((ext_vector_type(8))) float v8f;

#define IN_FEATS 384
#define N_HIDDEN 128
#define N_CLASSES 16

// ---------------------------------------------------------------- utilities
__global__ __launch_bounds__(256) void zero_kernel(float* __restrict__ p, long long n) {
    long long i = (long long)blockIdx.x * blockDim.x + threadIdx.x;
    long long stride = (long long)gridDim.x * blockDim.x;
    for (; i < n; i += stride) p[i] = 0.0f;
}

__global__ __launch_bounds__(256) void degree_kernel(const int* __restrict__ src,
                                                     const int* __restrict__ dst,
                                                     float* __restrict__ deg_out,
                                                     float* __restrict__ deg_in,
                                                     int nE) {
    int e = blockIdx.x * 256 + threadIdx.x;
    if (e >= nE) return;
    atomicAdd(&deg_out[src[e]], 1.0f);
    atomicAdd(&deg_in[dst[e]], 1.0f);
}

__global__ __launch_bounds__(256) void rsqrt_kernel(float* __restrict__ a,
                                                    float* __restrict__ b, int n) {
    int i = blockIdx.x * 256 + threadIdx.x;
    if (i >= n) return;
    float x = a[i];
    a[i] = (x > 0.0f) ? rsqrtf(x) : 0.0f;
    float y = b[i];
    b[i] = (y > 0.0f) ? rsqrtf(y) : 0.0f;
}

// -------------------------------------------------- row-scaled GEMM via WMMA
// out[N x F] = (H * ns[:,None]) @ W   (H: N x K row-major, W: K x F row-major)
// One wave computes a full 16 x F output slab (F = CT*16) so the A fragment
// is loaded once per K-step and reused across CT back-to-back WMMAs.
// Uses V_WMMA_F32_16X16X4_F32 (full fp32, matches the fp32 reference).
// Requirements: K multiple of 4, F == CT*16, rows handled in 16-row tiles.
template <int CT>
__global__ __launch_bounds__(64) void gemm_scaled_wmma(const float* __restrict__ H,
                                                       const float* __restrict__ ns,
                                                       const float* __restrict__ W,
                                                       float* __restrict__ out,
                                                       int K, int nRowTiles) {
    constexpr int F = CT * 16;
    const int wave = threadIdx.x >> 5;
    const int lane = threadIdx.x & 31;
    const int m  = lane & 15;   // A-tile row / B,C,D-tile column
    const int hi = lane >> 4;   // lane-half selector

    const int rowTile = blockIdx.x * 2 + wave;
    if (rowTile >= nRowTiles) return;       // wave-uniform: EXEC stays all-ones
    const int row0 = rowTile * 16;

    const float scale = ns[row0 + m];
    const float* __restrict__ arow = H + (size_t)(row0 + m) * K;
    const float* __restrict__ bbase = W + m;   // + ka*F + c*16

    v8f acc[CT];
#pragma unroll
    for (int c = 0; c < CT; ++c) acc[c] = (v8f){};

#pragma unroll 2
    for (int k0 = 0; k0 < K; k0 += 4) {
        const int ka = k0 + 2 * hi;
        // A fragment (16x4 fp32): lane holds A[m][ka], A[m][ka+1], scaled once.
        float2 t = *(const float2*)(arow + ka);
        v2f a;
        a.x = t.x * scale;
        a.y = t.y * scale;
        const float* __restrict__ brow = bbase + (size_t)ka * F;
#pragma unroll
        for (int c = 0; c < CT; ++c) {
            // B fragment (4x16 fp32): lane holds W[ka][c*16+m], W[ka+1][c*16+m]
            v2f b;
            b.x = brow[c * 16];
            b.y = brow[c * 16 + F];
            acc[c] = __builtin_amdgcn_wmma_f32_16x16x4_f32(false, a, false, b,
                                                           (short)0, acc[c], false, false);
        }
    }

    // D layout: VGPR v -> row = row0 + v + 8*hi, col = c*16 + m
    float* __restrict__ obase = out + (size_t)(row0 + 8 * hi) * F + m;
#pragma unroll
    for (int c = 0; c < CT; ++c) {
#pragma unroll
        for (int v = 0; v < 8; ++v)
            obase[(size_t)v * F + c * 16] = acc[c][v];
    }
}

// ------------------------------------------------------ edge scatter-add
// agg[dst[e]] += hw[src[e]]  (per 4-float group; F power of two, >= 16)
__global__ __launch_bounds__(256) void scatter_kernel(const int* __restrict__ src,
                                                      const int* __restrict__ dst,
                                                      const float* __restrict__ hw,
                                                      float* __restrict__ agg,
                                                      int nEdges, int F, int lgF4) {
    const int F4 = F >> 2;
    const int tid = blockIdx.x * 256 + threadIdx.x;
    const int e = tid >> lgF4;
    const int j = tid & (F4 - 1);
    if (e >= nEdges) return;
    const int s = src[e];
    const int d = dst[e];
    const float4 v = ((const float4*)(hw + (size_t)s * F))[j];
    float* o = agg + (size_t)d * F + 4 * j;
    atomicAdd(o + 0, v.x);
    atomicAdd(o + 1, v.y);
    atomicAdd(o + 2, v.z);
    atomicAdd(o + 3, v.w);
}

// ---------------------------------------------- out = agg*nd[:,None] + b (+relu)
__global__ __launch_bounds__(256) void finalize_kernel(const float* __restrict__ agg,
                                                       const float* __restrict__ nd,
                                                       const float* __restrict__ bias,
                                                       float* __restrict__ out,
                                                       int N, int lgF, int relu) {
    const int F = 1 << lgF;
    long long t = (long long)blockIdx.x * 256 + threadIdx.x;
    if (t >= ((long long)N << lgF)) return;
    const int i = (int)(t >> lgF);
    const int f = (int)(t & (F - 1));
    float v = agg[t] * nd[i] + bias[f];
    if (relu) v = fmaxf(v, 0.0f);
    out[t] = v;
}

// ----------------------------------------------------------------- driver
extern "C" void kernel_launch(void* const* d_in, const int* in_sizes, int n_in,
                              void* d_out, int out_size, void* d_ws, size_t ws_size,
                              hipStream_t stream) {
    const float* feat = (const float*)d_in[0];
    const int*   src  = (const int*)d_in[1];
    const int*   dst  = (const int*)d_in[2];
    const float* W1   = (const float*)d_in[3];
    const float* b1   = (const float*)d_in[4];
    const float* W2   = (const float*)d_in[5];
    const float* b2   = (const float*)d_in[6];
    const float* W3   = (const float*)d_in[7];
    const float* b3   = (const float*)d_in[8];
    float* out = (float*)d_out;

    const int N = in_sizes[0] / IN_FEATS;   // 100000 (multiple of 16)
    const int E = in_sizes[1];              // 1600000

    // Workspace layout (floats): ns[N] | nd[N] | bufA[N*128] | bufB[N*128]
    float* ns   = (float*)d_ws;
    float* nd   = ns + N;
    float* bufA = nd + N;
    float* bufB = bufA + (size_t)N * N_HIDDEN;

    const int rowTiles  = N / 16;            // 6250
    const int rowBlocks = (rowTiles + 1) / 2;

    // --- degree norms -----------------------------------------------------
    zero_kernel<<<(2 * N + 255) / 256, 256, 0, stream>>>(ns, 2LL * N);
    degree_kernel<<<(E + 255) / 256, 256, 0, stream>>>(src, dst, ns, nd, E);
    rsqrt_kernel<<<(N + 255) / 256, 256, 0, stream>>>(ns, nd, N);

    // --- layer 1: 384 -> 128 ---------------------------------------------
    {
        gemm_scaled_wmma<8><<<rowBlocks, 64, 0, stream>>>(feat, ns, W1, bufA, IN_FEATS, rowTiles);
        long long nAgg = (long long)N * N_HIDDEN;
        zero_kernel<<<(int)((nAgg + 255) / 256), 256, 0, stream>>>(bufB, nAgg);
        long long nScat = (long long)E * (N_HIDDEN / 4);
        scatter_kernel<<<(int)((nScat + 255) / 256), 256, 0, stream>>>(src, dst, bufA, bufB, E, N_HIDDEN, 5);
        finalize_kernel<<<(int)((nAgg + 255) / 256), 256, 0, stream>>>(bufB, nd, b1, bufA, N, 7, 1);
    }

    // --- layer 2: 128 -> 128 ---------------------------------------------
    {
        gemm_scaled_wmma<8><<<rowBlocks, 64, 0, stream>>>(bufA, ns, W2, bufB, N_HIDDEN, rowTiles);
        long long nAgg = (long long)N * N_HIDDEN;
        zero_kernel<<<(int)((nAgg + 255) / 256), 256, 0, stream>>>(bufA, nAgg);
        long long nScat = (long long)E * (N_HIDDEN / 4);
        scatter_kernel<<<(int)((nScat + 255) / 256), 256, 0, stream>>>(src, dst, bufB, bufA, E, N_HIDDEN, 5);
        finalize_kernel<<<(int)((nAgg + 255) / 256), 256, 0, stream>>>(bufA, nd, b2, bufB, N, 7, 1);
    }

    // --- layer 3: 128 -> 16 (no relu), scatter straight into d_out --------
    {
        gemm_scaled_wmma<1><<<rowBlocks, 64, 0, stream>>>(bufB, ns, W3, bufA, N_HIDDEN, rowTiles);
        long long nOut = (long long)N * N_CLASSES;
        zero_kernel<<<(int)((nOut + 255) / 256), 256, 0, stream>>>(out, nOut);
        long long nScat = (long long)E * (N_CLASSES / 4);
        scatter_kernel<<<(int)((nScat + 255) / 256), 256, 0, stream>>>(src, dst, bufA, out, E, N_CLASSES, 2);
        finalize_kernel<<<(int)((nOut + 255) / 256), 256, 0, stream>>>(out, nd, b3, out, N, 4, 0);
    }
}